// myGNN_12163347382982
// MI455X (gfx1250) — compile-verified
//
#include <hip/hip_runtime.h>
#include <math.h>

#define F_IN  57
#define KPAD1 64      // padded K for layer-1 GEMM
#define HID   128

typedef __attribute__((ext_vector_type(2))) float v2f;
typedef __attribute__((ext_vector_type(8))) float v8f;

// ---- order-preserving float<->uint encoding for atomic max over floats ----
__device__ __forceinline__ unsigned enc_f32(float x) {
    unsigned u = __float_as_uint(x);
    return (u & 0x80000000u) ? ~u : (u | 0x80000000u);
}
__device__ __forceinline__ float dec_f32(unsigned e) {
    unsigned u = (e & 0x80000000u) ? (e & 0x7fffffffu) : ~e;
    return __uint_as_float(u);
}

// ---- zero a region of workspace (grid-stride) ----
__global__ void k_zero(float* __restrict__ p, size_t n) {
    size_t i = (size_t)blockIdx.x * blockDim.x + threadIdx.x;
    size_t stride = (size_t)gridDim.x * blockDim.x;
    for (; i < n; i += stride) p[i] = 0.0f;
}

// ---- pad x[N,57] -> x_pad[N,64] (zero-filled tail columns) ----
__global__ void k_pad_x(const float* __restrict__ x, float* __restrict__ xp, int nNodes) {
    size_t i = (size_t)blockIdx.x * blockDim.x + threadIdx.x;
    size_t total = (size_t)nNodes * KPAD1;
    size_t stride = (size_t)gridDim.x * blockDim.x;
    for (; i < total; i += stride) {
        int n = (int)(i / KPAD1), k = (int)(i % KPAD1);
        xp[i] = (k < F_IN) ? x[(size_t)n * F_IN + k] : 0.0f;
    }
}

// ---- pack W[K,C] -> WT[C,Kpad] transposed + K zero-padded ----
__global__ void k_packT(const float* __restrict__ w, float* __restrict__ wt,
                        int K, int C, int Kpad) {
    int i = blockIdx.x * blockDim.x + threadIdx.x;
    int total = C * Kpad;
    if (i >= total) return;
    int c = i / Kpad, k = i % Kpad;
    wt[i] = (k < K) ? w[(size_t)k * C + c] : 0.0f;
}

// ---- GEMM1: h[N,128] = x_pad[N,64] @ WgT^T via V_WMMA_F32_16X16X4_F32 ----
// block = 256 = 8 waves; wave w owns 16-col tile w; blockIdx.x owns 16 rows.
// Fragment layouts (ISA 7.12.2): A/B lane l, slot v: K = k0 + v + 2*(l>>4);
// A: M = l&15, B: N = l&15. C/D: M = v + 8*(l>>4), N = l&15.
__global__ void k_gat_gemm1(const float* __restrict__ xp, const float* __restrict__ WgT,
                            float* __restrict__ h, int nNodes) {
    const int lane = threadIdx.x & 31;
    const int wv   = threadIdx.x >> 5;
    const int lh   = lane >> 4;
    const int lm   = lane & 15;
    const int m0   = blockIdx.x * 16;
    const int col  = wv * 16 + lm;
    const int rowA = min(m0 + lm, nNodes - 1);   // clamp: no divergent loads

    const float* ap = xp  + (size_t)rowA * KPAD1 + 2 * lh;
    const float* bp = WgT + (size_t)col  * KPAD1 + 2 * lh;

    v8f acc = {};
#pragma unroll
    for (int k0 = 0; k0 < KPAD1; k0 += 4) {
        const v2f a = *(const v2f*)(ap + k0);
        const v2f b = *(const v2f*)(bp + k0);
        acc = __builtin_amdgcn_wmma_f32_16x16x4_f32(false, a, false, b,
                                                    (short)0, acc, false, false);
    }
#pragma unroll
    for (int v = 0; v < 8; ++v) {
        const int row = m0 + v + 8 * lh;
        if (row < nNodes) h[(size_t)row * HID + col] = acc[v];
    }
}

// ---- per-node attention scalars: s[i]=h[i]·a_src, d[i]=h[i]·a_dst (1 wave / node) ----
__global__ void k_node_sd(const float* __restrict__ h, const float* __restrict__ a_s,
                          const float* __restrict__ a_d, float* __restrict__ s,
                          float* __restrict__ d, int nNodes) {
    const int lane = threadIdx.x & 31;
    const int node = blockIdx.x * 8 + (threadIdx.x >> 5);
    if (node >= nNodes) return;
    const float4 hv = ((const float4*)(h + (size_t)node * HID))[lane];
    const float4 u  = ((const float4*)a_s)[lane];
    const float4 v  = ((const float4*)a_d)[lane];
    float ps = hv.x * u.x + hv.y * u.y + hv.z * u.z + hv.w * u.w;
    float pd = hv.x * v.x + hv.y * v.y + hv.z * v.z + hv.w * v.w;
    for (int off = 16; off > 0; off >>= 1) {
        ps += __shfl_xor(ps, off, 32);
        pd += __shfl_xor(pd, off, 32);
    }
    if (lane == 0) { s[node] = ps; d[node] = pd; }
}

// ---- per-edge leaky-relu logit + segment max (1 thread / edge; self-loops appended) ----
__global__ void k_edge_logit(const int* __restrict__ ei, const float* __restrict__ s,
                             const float* __restrict__ d, float* __restrict__ logits,
                             unsigned* __restrict__ lmax, int E, int nNodes) {
    const int e = blockIdx.x * blockDim.x + threadIdx.x;
    const int total = E + nNodes;
    if (e >= total) return;
    int src, dst;
    if (e < E) { src = ei[e]; dst = ei[E + e]; } else { src = dst = e - E; }
    float lg = s[src] + d[dst];
    lg = lg > 0.0f ? lg : 0.2f * lg;
    logits[e] = lg;
    atomicMax(lmax + dst, enc_f32(lg));
}

// ---- per-edge exp + weighted scatter of h[src] into num[dst] (1 wave / edge) ----
__global__ void k_edge_accum(const int* __restrict__ ei, const float* __restrict__ logits,
                             const unsigned* __restrict__ lmax, const float* __restrict__ h,
                             float* __restrict__ denom, float* __restrict__ num,
                             int E, int nNodes) {
    const int lane = threadIdx.x & 31;
    const int e = blockIdx.x * 8 + (threadIdx.x >> 5);
    const int total = E + nNodes;
    if (e >= total) return;
    int src, dst;
    if (e < E) { src = ei[e]; dst = ei[E + e]; } else { src = dst = e - E; }
    const float w = expf(logits[e] - dec_f32(lmax[dst]));
    if (lane == 0) atomicAdd(denom + dst, w);
    const float4 hv = ((const float4*)(h + (size_t)src * HID))[lane];
    float* np = num + (size_t)dst * HID + lane * 4;
    atomicAdd(np + 0, w * hv.x);
    atomicAdd(np + 1, w * hv.y);
    atomicAdd(np + 2, w * hv.z);
    atomicAdd(np + 3, w * hv.w);
}

// ---- GAT epilogue: h1 = relu(num/denom + b_gat), in place over num ----
__global__ void k_gat_final(float* __restrict__ num, const float* __restrict__ denom,
                            const float* __restrict__ bgat, int nNodes) {
    const int lane = threadIdx.x & 31;
    const int node = blockIdx.x * 8 + (threadIdx.x >> 5);
    if (node >= nNodes) return;
    const float dn = denom[node];          // > 0 (self-loop guarantees it)
    float4* p = (float4*)(num + (size_t)node * HID);
    float4 v = p[lane];
    const float4 b = ((const float4*)bgat)[lane];
    v.x = fmaxf(v.x / dn + b.x, 0.0f);
    v.y = fmaxf(v.y / dn + b.y, 0.0f);
    v.z = fmaxf(v.z / dn + b.z, 0.0f);
    v.w = fmaxf(v.w / dn + b.w, 0.0f);
    p[lane] = v;
}

// ---- SAGE scatter: msum[dst] += h1[src]; cnt[dst] += 1 (1 wave / edge) ----
__global__ void k_sage_gather(const int* __restrict__ ei, const float* __restrict__ h1,
                              float* __restrict__ msum, float* __restrict__ cnt, int E) {
    const int lane = threadIdx.x & 31;
    const int e = blockIdx.x * 8 + (threadIdx.x >> 5);
    if (e >= E) return;
    const int src = ei[e], dst = ei[E + e];
    if (lane == 0) atomicAdd(cnt + dst, 1.0f);
    const float4 hv = ((const float4*)(h1 + (size_t)src * HID))[lane];
    float* mp = msum + (size_t)dst * HID + lane * 4;
    atomicAdd(mp + 0, hv.x);
    atomicAdd(mp + 1, hv.y);
    atomicAdd(mp + 2, hv.z);
    atomicAdd(mp + 3, hv.w);
}

// ---- mean = msum / max(cnt,1), in place ----
__global__ void k_sage_mean(float* __restrict__ msum, const float* __restrict__ cnt, int nNodes) {
    const int lane = threadIdx.x & 31;
    const int node = blockIdx.x * 8 + (threadIdx.x >> 5);
    if (node >= nNodes) return;
    const float inv = 1.0f / fmaxf(cnt[node], 1.0f);
    float4* p = (float4*)(msum + (size_t)node * HID);
    float4 v = p[lane];
    v.x *= inv; v.y *= inv; v.z *= inv; v.w *= inv;
    p[lane] = v;
}

// ---- GEMM2: h2 = relu(mean@Wl + h1@Wr + bl), dual-WMMA K-loop (K=128) ----
__global__ void k_sage_gemm2(const float* __restrict__ mean, const float* __restrict__ h1,
                             const float* __restrict__ WlT, const float* __restrict__ WrT,
                             const float* __restrict__ bl, float* __restrict__ h2, int nNodes) {
    const int lane = threadIdx.x & 31;
    const int wv   = threadIdx.x >> 5;
    const int lh   = lane >> 4;
    const int lm   = lane & 15;
    const int m0   = blockIdx.x * 16;
    const int col  = wv * 16 + lm;
    const int rowA = min(m0 + lm, nNodes - 1);   // clamp: no divergent loads

    const float* a1p = mean + (size_t)rowA * HID + 2 * lh;
    const float* a2p = h1   + (size_t)rowA * HID + 2 * lh;
    const float* b1p = WlT  + (size_t)col  * HID + 2 * lh;
    const float* b2p = WrT  + (size_t)col  * HID + 2 * lh;

    v8f acc = {};
#pragma unroll 8
    for (int k0 = 0; k0 < HID; k0 += 4) {
        const v2f a1 = *(const v2f*)(a1p + k0);
        const v2f b1 = *(const v2f*)(b1p + k0);
        acc = __builtin_amdgcn_wmma_f32_16x16x4_f32(false, a1, false, b1,
                                                    (short)0, acc, false, false);
        const v2f a2 = *(const v2f*)(a2p + k0);
        const v2f b2 = *(const v2f*)(b2p + k0);
        acc = __builtin_amdgcn_wmma_f32_16x16x4_f32(false, a2, false, b2,
                                                    (short)0, acc, false, false);
    }
    const float bias = bl[col];
#pragma unroll
    for (int v = 0; v < 8; ++v) {
        const int row = m0 + v + 8 * lh;
        if (row < nNodes) h2[(size_t)row * HID + col] = fmaxf(acc[v] + bias, 0.0f);
    }
}

// ---- output: sigmoid(h2 @ W_out + b_out) (1 wave / node) ----
__global__ void k_out(const float* __restrict__ h2, const float* __restrict__ Wout,
                      const float* __restrict__ bout, float* __restrict__ out, int nNodes) {
    const int lane = threadIdx.x & 31;
    const int node = blockIdx.x * 8 + (threadIdx.x >> 5);
    if (node >= nNodes) return;
    const float4 hv = ((const float4*)(h2 + (size_t)node * HID))[lane];
    const float4 w  = ((const float4*)Wout)[lane];
    float p = hv.x * w.x + hv.y * w.y + hv.z * w.z + hv.w * w.w;
    for (int off = 16; off > 0; off >>= 1) p += __shfl_xor(p, off, 32);
    if (lane == 0) {
        const float z = p + bout[0];
        out[node] = 1.0f / (1.0f + expf(-z));
    }
}

extern "C" void kernel_launch(void* const* d_in, const int* in_sizes, int n_in,
                              void* d_out, int out_size, void* d_ws, size_t ws_size,
                              hipStream_t stream) {
    const float* x    = (const float*)d_in[0];
    const int*   ei   = (const int*)  d_in[1];
    const float* Wg   = (const float*)d_in[2];
    const float* a_s  = (const float*)d_in[3];
    const float* a_d  = (const float*)d_in[4];
    const float* bg   = (const float*)d_in[5];
    const float* Wl   = (const float*)d_in[6];
    const float* bl   = (const float*)d_in[7];
    const float* Wr   = (const float*)d_in[8];
    const float* Wout = (const float*)d_in[9];
    const float* bout = (const float*)d_in[10];
    float* out = (float*)d_out;

    const int nNodes = in_sizes[0] / F_IN;
    const int E      = in_sizes[1] / 2;

    // workspace layout (floats). Zeroed region is [num .. lmax] (contiguous).
    float* ws = (float*)d_ws;
    size_t o = 0;
    float* h      = ws + o; o += (size_t)nNodes * HID;
    float* h2     = ws + o; o += (size_t)nNodes * HID;
    float* xp     = ws + o; o += (size_t)nNodes * KPAD1;
    float* WgT    = ws + o; o += (size_t)HID * KPAD1;
    float* WlT    = ws + o; o += (size_t)HID * HID;
    float* WrT    = ws + o; o += (size_t)HID * HID;
    float* s      = ws + o; o += nNodes;
    float* dvec   = ws + o; o += nNodes;
    float* logits = ws + o; o += (size_t)E + nNodes;
    float* num    = ws + o; o += (size_t)nNodes * HID;   // becomes h1 after k_gat_final
    float* msum   = ws + o; o += (size_t)nNodes * HID;   // becomes mean after k_sage_mean
    float* denom  = ws + o; o += nNodes;
    float* cnt    = ws + o; o += nNodes;
    unsigned* lmax = (unsigned*)(ws + o); o += nNodes;    // 0 encodes below -inf

    const size_t zeroWords = 2 * (size_t)nNodes * HID + 3 * (size_t)nNodes;

    k_zero<<<4096, 256, 0, stream>>>(num, zeroWords);
    k_pad_x<<<4096, 256, 0, stream>>>(x, xp, nNodes);
    k_packT<<<(HID * KPAD1 + 255) / 256, 256, 0, stream>>>(Wg, WgT, F_IN, HID, KPAD1);
    k_packT<<<(HID * HID + 255) / 256, 256, 0, stream>>>(Wl, WlT, HID, HID, HID);
    k_packT<<<(HID * HID + 255) / 256, 256, 0, stream>>>(Wr, WrT, HID, HID, HID);

    k_gat_gemm1<<<(nNodes + 15) / 16, 256, 0, stream>>>(xp, WgT, h, nNodes);
    k_node_sd<<<(nNodes + 7) / 8, 256, 0, stream>>>(h, a_s, a_d, s, dvec, nNodes);
    k_edge_logit<<<(E + nNodes + 255) / 256, 256, 0, stream>>>(ei, s, dvec, logits, lmax, E, nNodes);
    k_edge_accum<<<(E + nNodes + 7) / 8, 256, 0, stream>>>(ei, logits, lmax, h, denom, num, E, nNodes);
    k_gat_final<<<(nNodes + 7) / 8, 256, 0, stream>>>(num, denom, bg, nNodes);   // num -> h1

    k_sage_gather<<<(E + 7) / 8, 256, 0, stream>>>(ei, num, msum, cnt, E);
    k_sage_mean<<<(nNodes + 7) / 8, 256, 0, stream>>>(msum, cnt, nNodes);        // msum -> mean
    k_sage_gemm2<<<(nNodes + 15) / 16, 256, 0, stream>>>(msum, num, WlT, WrT, bl, h2, nNodes);

    k_out<<<(nNodes + 7) / 8, 256, 0, stream>>>(h2, Wout, bout, out, nNodes);
}